// CCP_8873402433933
// MI455X (gfx1250) — compile-verified
//
#include <hip/hip_runtime.h>
#include <stdint.h>

// ---------------------------------------------------------------------------
// NCD-via-LZW for MI455X (gfx1250).
//
// Shape constants from the reference.
#define BATCH   32
#define CH      3
#define LVL     8          // alphabet
#define PMAPS   16
#define MLEN    1024
#define NLEN    1024

#define STR_BYTES (BATCH * CH * NLEN)        // 98304  quantized strings (u8)
#define PMP_BYTES (CH * PMAPS * MLEN)        // 49152  pmaps narrowed to u8
#define OFF_STR   0
#define OFF_PMP   STR_BYTES
#define OFF_CNT   (STR_BYTES + PMP_BYTES)    // int32 counts: [96 c_s][48 c_p][1536 c_sp]

#define N_CS      (BATCH * CH)               // 96
#define N_CP      (CH * PMAPS)               // 48
#define N_CSP     (BATCH * CH * PMAPS)       // 1536
#define N_TASK    (N_CS + N_CP + N_CSP)      // 1680 independent LZW chains

// One LZW task per wave32. Trie: int16 trans[DICT_ROWS][8] in LDS.
// Worst case nfree = 8 + 2047 = 2055  ->  2056 rows; each row = 16 bytes.
#define WAVES_PER_BLOCK 4
#define DICT_ROWS  2056
#define TRIE_BYTES (DICT_ROWS * 8 * 2)       // 32896
#define SEQ_BYTES  2048
#define WAVE_LDS   (TRIE_BYTES + SEQ_BYTES)  // 34944 (16B aligned)
// Block dynamic LDS = 4 * 34944 = 139776 B -> 2 blocks per 320KB WGP.

// ---- CDNA5 async global->LDS staging (gfx1250-specific path) --------------
__device__ __forceinline__ void async_ld_lds_b128(uint32_t lds_off, uint64_t gaddr) {
    // VDST = per-lane LDS byte offset, VADDR = per-lane 64-bit global address.
    asm volatile("global_load_async_to_lds_b128 %0, %1, off"
                 :: "v"(lds_off), "v"(gaddr)
                 : "memory");
}
__device__ __forceinline__ void wait_async0() {
    asm volatile("s_wait_asynccnt 0x0" ::: "memory");
}

// ---------------------------------------------------------------------------
// Kernel 1: quantize x along the space-filling curve (first-min argmin, same
// tie behavior as jnp.argmin) + narrow pmaps int32 -> u8.
__global__ __launch_bounds__(256)
void quantize_kernel(const float* __restrict__ x,
                     const int*   __restrict__ curve,
                     const float* __restrict__ levels,
                     const int*   __restrict__ pmaps,
                     uint8_t*     __restrict__ ws) {
    int j = blockIdx.x * blockDim.x + threadIdx.x;
    if (j < STR_BYTES) {
        int n  = j & (NLEN - 1);
        int bc = j >> 10;                // b*CH + c
        int c  = bc % CH;
        float v = x[bc * NLEN + curve[n]];
        const float* lv = levels + c * LVL;
        float best = fabsf(v - lv[0]);
        int idx = 0;
#pragma unroll
        for (int l = 1; l < LVL; ++l) {
            float d = fabsf(v - lv[l]);
            if (d < best) { best = d; idx = l; }   // strict '<' -> first min
        }
        ws[OFF_STR + j] = (uint8_t)idx;
    } else if (j < STR_BYTES + PMP_BYTES) {
        int k = j - STR_BYTES;
        ws[OFF_PMP + k] = (uint8_t)pmaps[k];       // values already in [0,8)
    }
}

// ---------------------------------------------------------------------------
// Branchless LZW step, mirroring the reference's unconditional
// trans[cur,c] = hit ? nxt : nfree (hit-store is a semantic no-op).
// Only the trie load sits on the loop-carried dependence chain.
#define LZW_STEP(c_)                                            \
    {                                                           \
        int c__  = (int)(c_);                                   \
        int idx  = (cur << 3) | c__;                            \
        int nxt  = (int)trie[idx];      /* dependent LDS load */\
        bool hit = (nxt >= 0);                                  \
        trie[idx] = (int16_t)(hit ? nxt : nfree);               \
        cur   = hit ? nxt : c__;                                \
        nfree += hit ? 0 : 1;                                   \
        cnt   += hit ? 0 : 1;                                   \
    }

// ---------------------------------------------------------------------------
// Kernel 2: one LZW chain per wave. Trie + sequence live entirely in LDS.
__global__ __launch_bounds__(WAVES_PER_BLOCK * 32)
void lzw_kernel(const uint8_t* __restrict__ ws, int* __restrict__ counts) {
    extern __shared__ unsigned char smem[];
    const int lane = threadIdx.x & 31;
    const int w    = threadIdx.x >> 5;
    const int task = blockIdx.x * WAVES_PER_BLOCK + w;
    if (task >= N_TASK) return;

    // Workgroup-relative LDS byte offset of this wave's region (low 32 bits of
    // the generic shared pointer are the LDS offset on gfx1250 flat aperture).
    const uint32_t lds_base = (uint32_t)(uintptr_t)(&smem[0]) + (uint32_t)(w * WAVE_LDS);
    const uint32_t seq_lds  = lds_base + TRIE_BYTES;

    // ---- task decode ----
    const uint8_t* seg0;
    const uint8_t* seg1 = nullptr;
    int len;
    int* out_ptr;
    if (task < N_CS) {                               // c_s[b,c]
        seg0 = ws + OFF_STR + task * NLEN;  len = NLEN;
        out_ptr = counts + task;
    } else if (task < N_CS + N_CP) {                 // c_p[c,p]
        int t = task - N_CS;
        seg0 = ws + OFF_PMP + t * MLEN;     len = MLEN;
        out_ptr = counts + N_CS + t;
    } else {                                         // c_sp[b,c,p]
        int t = task - (N_CS + N_CP);
        int b = t / (CH * PMAPS);
        int r = t % (CH * PMAPS);                    // r = c*16 + p
        seg0 = ws + OFF_STR + (b * CH + r / PMAPS) * NLEN;
        seg1 = ws + OFF_PMP + r * MLEN;
        len  = NLEN + MLEN;
        out_ptr = counts + N_CS + N_CP + t;
    }

    // ---- stage sequence global->LDS with async b128 copies (512 B / issue) ----
    {
        uint32_t l0 = seq_lds + (uint32_t)(lane * 16);
        uint64_t g0 = (uint64_t)(uintptr_t)seg0 + (uint64_t)(lane * 16);
        async_ld_lds_b128(l0,       g0);
        async_ld_lds_b128(l0 + 512, g0 + 512);
        if (seg1) {
            uint32_t l1 = l0 + 1024;
            uint64_t g1 = (uint64_t)(uintptr_t)seg1 + (uint64_t)(lane * 16);
            async_ld_lds_b128(l1,       g1);
            async_ld_lds_b128(l1 + 512, g1 + 512);
        }
    }

    // ---- init trie rows [0, 8+len) to -1: one 16B row per ds_store_b128 ----
    {
        uint4* t128 = (uint4*)(smem + (size_t)w * WAVE_LDS);
        const uint4 m1 = make_uint4(0xFFFFFFFFu, 0xFFFFFFFFu, 0xFFFFFFFFu, 0xFFFFFFFFu);
        const int rows = LVL + len;                  // max nfree+1 rows needed
        for (int i = lane; i < rows; i += 32) t128[i] = m1;
    }

    wait_async0();   // sequence resident in LDS

    // ---- serial trie walk on lane 0 (loop-carried dependence) ----
    if (lane == 0) {
        int16_t*        trie  = (int16_t*)(smem + (size_t)w * WAVE_LDS);
        const uint32_t* seq32 = (const uint32_t*)(smem + (size_t)w * WAVE_LDS + TRIE_BYTES);

        uint32_t word = seq32[0];
        int cur   = (int)(word & 7u);
        int nfree = LVL;
        int cnt   = 0;
        // symbols 1..3 of the first word
        LZW_STEP((word >>  8) & 7u);
        LZW_STEP((word >> 16) & 7u);
        LZW_STEP((word >> 24) & 7u);
        // remaining words, 4 symbols each (len is a multiple of 4)
        for (int i4 = 4; i4 < len; i4 += 4) {
            word = seq32[i4 >> 2];
            LZW_STEP( word        & 7u);
            LZW_STEP((word >>  8) & 7u);
            LZW_STEP((word >> 16) & 7u);
            LZW_STEP((word >> 24) & 7u);
        }
        *out_ptr = cnt + 1;                          // emit final pending phrase
    }
}

// ---------------------------------------------------------------------------
// Kernel 3: NCD = (c_sp - min(c_s,c_p)) / max(c_s,c_p), out [B, C*P] flat.
__global__ __launch_bounds__(256)
void ncd_kernel(const int* __restrict__ counts, float* __restrict__ out) {
    int t = blockIdx.x * blockDim.x + threadIdx.x;
    if (t >= N_CSP) return;
    int b = t / (CH * PMAPS);
    int r = t % (CH * PMAPS);                        // c*16 + p
    int c = r / PMAPS;
    float cs  = (float)counts[b * CH + c];
    float cp  = (float)counts[N_CS + r];
    float csp = (float)counts[N_CS + N_CP + t];
    out[t] = (csp - fminf(cs, cp)) / fmaxf(cs, cp);
}

// ---------------------------------------------------------------------------
extern "C" void kernel_launch(void* const* d_in, const int* in_sizes, int n_in,
                              void* d_out, int out_size, void* d_ws, size_t ws_size,
                              hipStream_t stream) {
    const float* x      = (const float*)d_in[0];
    const int*   curve  = (const int*)d_in[1];
    const float* levels = (const float*)d_in[2];
    const int*   pmaps  = (const int*)d_in[3];
    // d_in[4] == i, unused by the reference.
    uint8_t* ws  = (uint8_t*)d_ws;
    float*   out = (float*)d_out;
    (void)in_sizes; (void)n_in; (void)out_size; (void)ws_size;

    // 1) quantize + narrow
    {
        int total = STR_BYTES + PMP_BYTES;           // 147456
        quantize_kernel<<<(total + 255) / 256, 256, 0, stream>>>(x, curve, levels, pmaps, ws);
    }
    // 2) 1680 LZW chains, one per wave32, trie+seq in LDS
    {
        int blocks = (N_TASK + WAVES_PER_BLOCK - 1) / WAVES_PER_BLOCK;  // 420
        size_t lds = (size_t)WAVES_PER_BLOCK * WAVE_LDS;                // 139776 B
        lzw_kernel<<<blocks, WAVES_PER_BLOCK * 32, lds, stream>>>(ws, (int*)(ws + OFF_CNT));
    }
    // 3) NCD epilogue
    ncd_kernel<<<(N_CSP + 255) / 256, 256, 0, stream>>>((const int*)(ws + OFF_CNT), out);
}